// WFSA_24773371364104
// MI455X (gfx1250) — compile-verified
//
#include <hip/hip_runtime.h>

typedef __attribute__((ext_vector_type(2))) float v2f;
typedef __attribute__((ext_vector_type(8))) float v8f;
typedef __attribute__((ext_vector_type(4))) unsigned int u32x4;
typedef __attribute__((ext_vector_type(8))) int i32x8;
typedef __attribute__((ext_vector_type(4))) int i32x4;

#define WX 128               // n_states
#define WY 64                // n_symbols
#define WT 8192              // sequence length
#define NCHUNK 128
#define CLEN (WT / NCHUNK)   // 64 timesteps per chunk
#define PSTRIDE 132          // padded row stride for P in LDS (A-frag bank spread)

// B-fragment layout for M matrices: frag[p][j] = {M[2p][j], M[2p+1][j]}, p=0..63.
// Pair-row stride padded 128->144 so half-wave 1 (+144 pairs = 288 words % 64 = 32)
// lands on banks 32..63 while half-wave 0 uses banks 0..31: conflict-free b64 loads.
#define MPSTRIDE 144
#define MSZ (64 * MPSTRIDE * 2)   // floats per symbol matrix buffer (18432 = 72 KB)

#ifndef __has_builtin
#define __has_builtin(x) 0
#endif
#if defined(__HIP_DEVICE_COMPILE__) && \
    __has_builtin(__builtin_amdgcn_tensor_load_to_lds) && \
    __has_builtin(__builtin_amdgcn_s_wait_tensorcnt)
#define USE_TDM 1
#else
#define USE_TDM 0
#endif

// ---------------------------------------------------------------------------
// Kernel 1: Aeff[s] in B-fragment layout.
// element (k, j) of M_s  ->  Aeff_s[ ((k>>1)*MPSTRIDE + j)*2 + (k&1) ]
// block -> (s, k), thread -> j.   M_s[k][j] = sum_u A[k][u][j] * phi[s][u]
// ---------------------------------------------------------------------------
__global__ __launch_bounds__(WX) void wfsa_aeff(const float* __restrict__ A,
                                                const float* __restrict__ phi,
                                                float* __restrict__ Aeff) {
    __shared__ float prow[WY];
    const int s = blockIdx.x >> 7;
    const int k = blockIdx.x & (WX - 1);
    const int j = threadIdx.x;
    if (j < WY) prow[j] = phi[s * WY + j];
    __syncthreads();
    float acc = 0.0f;
    const float* Ak = A + (size_t)k * WY * WX;
    #pragma unroll 8
    for (int u = 0; u < WY; ++u)
        acc = fmaf(Ak[u * WX + j], prow[u], acc);
    float* dst = Aeff + (size_t)s * MSZ;
    dst[(((k >> 1) * MPSTRIDE + j) << 1) | (k & 1)] = acc;
    // zero the pad columns (j = 128..143) once per pair-row (even-k blocks)
    if ((k & 1) == 0 && j < 32)
        dst[(((k >> 1) * MPSTRIDE + WX + (j >> 1)) << 1) | (j & 1)] = 0.0f;
}

#if USE_TDM
// One 2D TDM descriptor: dense copy of a 288x64 f32 region (72 KB) global -> LDS.
// D# layout per CDNA5 ISA 8.3/8.4. Groups 2/3 zero (2D tensor).
__device__ __forceinline__ void tdm_load_matrix(const float* gsrc, float* lds_dst) {
    const unsigned long long ga = (unsigned long long)(uintptr_t)gsrc;
    const unsigned lds_off = (unsigned)(uintptr_t)lds_dst;  // LDS aperture: low 32 bits = byte offset
    const int D0 = 2 * MPSTRIDE;   // 288 floats per pair-row line
    const int D1 = 64;             // 64 pair-rows
    u32x4 g0;
    g0[0] = 1u;                                             // count=1, user descriptor
    g0[1] = lds_off;                                        // lds_addr (bytes)
    g0[2] = (unsigned)ga;                                   // global_addr[31:0]
    g0[3] = ((unsigned)(ga >> 32) & 0x01FFFFFFu)            // global_addr[56:32]
            | 0x80000000u;                                  // type=2 ("image") in bits[127:126]
    i32x8 g1;
    g1[0] = 0x00020000;      // workgroup_mask=0, data_size=2 (4 bytes), no barrier/iter/pad
    g1[1] = (D0 << 16);      // tensor_dim0[15:0] (bits 63:48)
    g1[2] = (D1 << 16);      // tensor_dim0[31:16]=0, tensor_dim1[15:0] (bits 95:80)
    g1[3] = (D0 << 16);      // tensor_dim1[31:16]=0, tile_dim0 (bits 127:112)
    g1[4] = D1;              // tile_dim1 (bits 143:128), tile_dim2=0
    g1[5] = D0;              // tensor_dim0_stride[31:0] (elements)
    g1[6] = 0;               // tensor_dim0_stride[47:32]=0, tensor_dim1_stride lo=0
    g1[7] = 0;
    const i32x4 gz4 = {0, 0, 0, 0};
    const i32x8 gz8 = {0, 0, 0, 0, 0, 0, 0, 0};
    __builtin_amdgcn_tensor_load_to_lds(g0, g1, gz4, gz4, gz8, 0);
}
#endif

// ---------------------------------------------------------------------------
// Kernel 2: per-chunk matrix chain product, FP32 WMMA + TDM double buffer.
// Chunk c computes P_c = M_{c*CLEN} * M_{c*CLEN+1} * ... (left-to-right).
// 256 threads = 8 waves; wave w owns output rows [16w, 16w+16).
// ---------------------------------------------------------------------------
__global__ __launch_bounds__(256) void wfsa_chunk(const int* __restrict__ inp,
                                                  const float* __restrict__ Aeff,
                                                  float* __restrict__ ChunkP) {
    extern __shared__ float smem[];
    float* Pbuf  = smem;                     // WX rows * PSTRIDE floats
    float* Mbuf0 = smem + WX * PSTRIDE;      // two MSZ fragment-layout M buffers
    float* Mbuf1 = Mbuf0 + MSZ;

    const int tid  = threadIdx.x;
    const int wave = tid >> 5;
    const int lane = tid & 31;
    const int half = lane >> 4;
    const int l16  = lane & 15;
    const int c    = blockIdx.x;
    const int row0 = wave * 16;

    // P <- identity
    for (int i = tid; i < WX * WX; i += 256) {
        int r = i >> 7, col = i & (WX - 1);
        Pbuf[r * PSTRIDE + col] = (r == col) ? 1.0f : 0.0f;
    }

    // preload M_{t0} into buffer 0
    const float* M0 = Aeff + (size_t)inp[c * CLEN] * MSZ;
#if USE_TDM
    if (wave == 0) {
        tdm_load_matrix(M0, Mbuf0);
        __builtin_amdgcn_s_wait_tensorcnt(0);
    }
    __syncthreads();
#else
    __syncthreads();
    {
        const float4* src = reinterpret_cast<const float4*>(M0);
        float4*       dst = reinterpret_cast<float4*>(Mbuf0);
        for (int i = tid; i < MSZ / 4; i += 256) dst[i] = src[i];
    }
    __syncthreads();
#endif

    for (int it = 0; it < CLEN; ++it) {
        const int t = c * CLEN + it;
        float* cur = (it & 1) ? Mbuf1 : Mbuf0;
        float* nxt = (it & 1) ? Mbuf0 : Mbuf1;

#if USE_TDM
        // overlap: DMA M_{t+1} into the other buffer while we compute on cur
        if (it + 1 < CLEN && wave == 0) {
            const float* Mn = Aeff + (size_t)inp[t + 1] * MSZ;
            tdm_load_matrix(Mn, nxt);
        }
#else
        if (it + 1 < CLEN) {
            const float* Mn = Aeff + (size_t)inp[t + 1] * MSZ;
            const float4* src = reinterpret_cast<const float4*>(Mn);
            float4*       dst = reinterpret_cast<float4*>(nxt);
            for (int i = tid; i < MSZ / 4; i += 256) dst[i] = src[i];
        }
#endif

        // Pnew[rows of this wave] = P[rows of this wave] @ M_t
        const float* prow  = &Pbuf[(row0 + l16) * PSTRIDE];
        const v2f*   bfrag = reinterpret_cast<const v2f*>(cur);  // [pair][MPSTRIDE] b64 frags
        v8f acc[8] = {};
        #pragma unroll 4
        for (int k0 = 0; k0 < WX; k0 += 4) {
            // A frag: row l16, K pair {k0+2h, k0+2h+1} -> one aligned b64 load
            v2f a = *reinterpret_cast<const v2f*>(&prow[k0 + 2 * half]);
            const int pbase = ((k0 >> 1) + half) * MPSTRIDE + l16;
            // gather all 8 B frags into independent regs first: lets multiple
            // ds_load_2addr_b64 stay in flight instead of wait-0 per WMMA pair
            v2f barr[8];
            #pragma unroll
            for (int nt = 0; nt < 8; ++nt) barr[nt] = bfrag[pbase + nt * 16];
            #pragma unroll
            for (int nt = 0; nt < 8; ++nt)
                acc[nt] = __builtin_amdgcn_wmma_f32_16x16x4_f32(
                    false, a, false, barr[nt], (short)0, acc[nt], false, false);
        }
        // write D tiles back into our own P rows (C/D layout: VGPR v -> row v+8*half)
        #pragma unroll
        for (int nt = 0; nt < 8; ++nt)
            #pragma unroll
            for (int v = 0; v < 8; ++v)
                Pbuf[(row0 + v + 8 * half) * PSTRIDE + nt * 16 + l16] = acc[nt][v];

#if USE_TDM
        if (wave == 0) __builtin_amdgcn_s_wait_tensorcnt(0);  // next buffer landed
#endif
        __syncthreads();
    }

    // P_c -> global (dense row-major)
    float* out = ChunkP + (size_t)c * (WX * WX);
    for (int i = tid; i < WX * WX; i += 256) {
        int r = i >> 7, col = i & (WX - 1);
        out[i] = Pbuf[r * PSTRIDE + col];
    }
}

// ---------------------------------------------------------------------------
// Kernel 3: x = final; for c = NCHUNK-1..0: x = P_c @ x;  out = init . x
// ---------------------------------------------------------------------------
__global__ __launch_bounds__(WX) void wfsa_combine(const float* __restrict__ ChunkP,
                                                   const float* __restrict__ initv,
                                                   const float* __restrict__ finalv,
                                                   float* __restrict__ out) {
    __shared__ float xv[WX];
    __shared__ float red[WX];
    const int i = threadIdx.x;
    xv[i] = finalv[i];
    __syncthreads();
    for (int c = NCHUNK - 1; c >= 0; --c) {
        const float* P = ChunkP + (size_t)c * (WX * WX) + (size_t)i * WX;
        float acc = 0.0f;
        #pragma unroll 8
        for (int j = 0; j < WX; ++j) acc = fmaf(P[j], xv[j], acc);
        __syncthreads();
        xv[i] = acc;
        __syncthreads();
    }
    red[i] = initv[i] * xv[i];
    __syncthreads();
    for (int s = WX / 2; s > 0; s >>= 1) {
        if (i < s) red[i] += red[i + s];
        __syncthreads();
    }
    if (i == 0) out[0] = red[0];
}

// ---------------------------------------------------------------------------
extern "C" void kernel_launch(void* const* d_in, const int* in_sizes, int n_in,
                              void* d_out, int out_size, void* d_ws, size_t ws_size,
                              hipStream_t stream) {
    const int*   inp   = (const int*)d_in[0];     // [T] symbol ids
    const float* A     = (const float*)d_in[1];   // [X,Y,X]
    const float* phi   = (const float*)d_in[2];   // [Y,Y]
    const float* initv = (const float*)d_in[3];   // [X]
    const float* finv  = (const float*)d_in[4];   // [X]
    float*       out   = (float*)d_out;

    float* Aeff   = (float*)d_ws;                  // WY * MSZ floats (4.5 MB, frag layout)
    float* ChunkP = Aeff + (size_t)WY * MSZ;       // NCHUNK*128*128 floats (8 MB)

    wfsa_aeff<<<WY * WX, WX, 0, stream>>>(A, phi, Aeff);

    const size_t shmem = (WX * PSTRIDE + 2 * MSZ) * sizeof(float); // ~210 KB
    wfsa_chunk<<<NCHUNK, 256, shmem, stream>>>(inp, Aeff, ChunkP);

    wfsa_combine<<<1, WX, 0, stream>>>(ChunkP, initv, finv, out);
}